// SSLEncoder_25967372272023
// MI455X (gfx1250) — compile-verified
//
#include <hip/hip_runtime.h>
#include <hip/hip_bf16.h>
#include <math.h>

#define N_NODES 50000
#define N_EDGES 800000
#define HDIM 128
#define NDIM 32
#define EDIM 4
#define NLAYERS 3
#define LN_EPS 1e-5f

typedef __attribute__((ext_vector_type(2))) float v2f;
typedef __attribute__((ext_vector_type(8))) float v8f;

// Native non-returning fp32 global atomic add, device scope (visible to the
// following kernel through L2). Avoids any CAS-loop expansion.
__device__ __forceinline__ void atomic_add_f32_dev(float* p, float v)
{
    asm volatile("global_atomic_add_f32 %0, %1, off scope:SCOPE_DEV"
                 :: "v"(p), "v"(v) : "memory");
}

// ---------------------------------------------------------------------------
// GEMM: C[M x 128] = A[M x K] @ W[K x 128] + bias[128], fp32 via
// V_WMMA_F32_16X16X4_F32. Wt is the COLUMN-MAJOR weight (Wt[col*K + k]) so
// each B fragment is one aligned 8-byte load. One block = 16 rows of A,
// 8 waves cover the 128 output columns. M multiple of 16 (50000 = 3125*16).
// A tile staged in LDS with pad=2 to avoid bank conflicts.
// ---------------------------------------------------------------------------
__global__ __launch_bounds__(256) void gemm_wmma_f32(
    const float* __restrict__ A, const float* __restrict__ Wt,
    const float* __restrict__ bias, float* __restrict__ C, int K)
{
    __shared__ float Atile[16 * (HDIM + 2)];
    const int ldA  = K + 2;
    const int row0 = blockIdx.x * 16;
    const int tid  = threadIdx.x;

    for (int i = tid; i < 16 * K; i += 256) {
        int r = i / K;
        int c = i - r * K;
        Atile[r * ldA + c] = A[(size_t)(row0 + r) * K + c];
    }
    __syncthreads();

    const int lane = tid & 31;
    const int wv   = tid >> 5;
    const int m    = lane & 15;   // A row / B,D col within tile
    const int kh   = lane >> 4;   // K-half select (f32 16x4 layout)
    const int col  = wv * 16 + m;

    const float* Wcol = Wt + (size_t)col * K;

    v8f acc = {};
    for (int k0 = 0; k0 < K; k0 += 4) {
        const int ka = k0 + 2 * kh;            // even -> 8B aligned
        const float2 av = *(const float2*)(Atile + m * ldA + ka);
        const float2 bv = *(const float2*)(Wcol + ka);
        v2f a, b;
        a.x = av.x; a.y = av.y;
        b.x = bv.x; b.y = bv.y;
        acc = __builtin_amdgcn_wmma_f32_16x16x4_f32(
            false, a, false, b, (short)0, acc, false, false);
    }

    const float bb = bias[col];
    #pragma unroll
    for (int r = 0; r < 8; ++r) {
        const int row = row0 + r + 8 * kh;     // D: M = r + 8*kh, N = lane&15
        C[(size_t)row * HDIM + col] = acc[r] + bb;
    }
}

// ---------------------------------------------------------------------------
// Weight fusion (tiny): WfusedT[l][j][k] = (Wn[l] @ Wm[l,:H])^T stored
// column-major for the GEMM's B-operand. Wefused = We @ Wm[H:]. Fused biases.
// ---------------------------------------------------------------------------
__global__ void fuse_node_weights(const float* __restrict__ Wn,
                                  const float* __restrict__ Wm,
                                  float* __restrict__ WfusedT)
{
    int idx = blockIdx.x * blockDim.x + threadIdx.x;
    if (idx >= NLAYERS * HDIM * HDIM) return;
    int l = idx / (HDIM * HDIM);
    int r = idx - l * HDIM * HDIM;
    int i = r / HDIM, j = r - (r / HDIM) * HDIM;   // i = K index, j = out col
    const float* wn  = Wn + (size_t)l * HDIM * HDIM + (size_t)i * HDIM;
    const float* wm1 = Wm + (size_t)l * 2 * HDIM * HDIM;   // rows 0..H-1
    float s = 0.f;
    for (int k = 0; k < HDIM; ++k) s += wn[k] * wm1[(size_t)k * HDIM + j];
    // transposed store: [l][j][i]
    WfusedT[((size_t)l * HDIM + j) * HDIM + i] = s;
}

__global__ void transpose_wp(const float* __restrict__ Wp,
                             float* __restrict__ WpT)
{
    int idx = blockIdx.x * blockDim.x + threadIdx.x;   // ND*H threads
    if (idx >= NDIM * HDIM) return;
    int i = idx / HDIM, j = idx - (idx / HDIM) * HDIM;
    WpT[(size_t)j * NDIM + i] = Wp[idx];
}

__global__ void fuse_edge_weights(const float* __restrict__ We,
                                  const float* __restrict__ Wm,
                                  float* __restrict__ Wefused)
{
    int idx = blockIdx.x * blockDim.x + threadIdx.x;
    if (idx >= NLAYERS * EDIM * HDIM) return;
    int l = idx / (EDIM * HDIM);
    int r = idx - l * EDIM * HDIM;
    int i = r / HDIM, j = r - (r / HDIM) * HDIM;
    const float* we  = We + ((size_t)l * EDIM + i) * HDIM;
    const float* wm2 = Wm + (size_t)l * 2 * HDIM * HDIM + (size_t)HDIM * HDIM;
    float s = 0.f;
    for (int k = 0; k < HDIM; ++k) s += we[k] * wm2[(size_t)k * HDIM + j];
    Wefused[idx] = s;
}

__global__ void fuse_biases(const float* __restrict__ bn,
                            const float* __restrict__ be,
                            const float* __restrict__ bm,
                            const float* __restrict__ Wm,
                            float* __restrict__ bfused,
                            float* __restrict__ befused)
{
    int idx = blockIdx.x * blockDim.x + threadIdx.x;
    if (idx >= NLAYERS * HDIM) return;
    int l = idx / HDIM, j = idx - (idx / HDIM) * HDIM;
    const float* wm1 = Wm + (size_t)l * 2 * HDIM * HDIM;
    const float* wm2 = wm1 + (size_t)HDIM * HDIM;
    float s1 = 0.f, s2 = 0.f;
    for (int k = 0; k < HDIM; ++k) {
        s1 += bn[l * HDIM + k] * wm1[(size_t)k * HDIM + j];
        s2 += be[l * HDIM + k] * wm2[(size_t)k * HDIM + j];
    }
    bfused[idx]  = s1;
    befused[idx] = s2 + bm[idx];
}

__global__ void zero_f32v4(float4* __restrict__ p, int n4)
{
    int i = blockIdx.x * blockDim.x + threadIdx.x;
    if (i < n4) p[i] = make_float4(0.f, 0.f, 0.f, 0.f);
}

// ---------------------------------------------------------------------------
// Edge kernel: one wave per edge. msg = m1[src] + edge_attr @ Wef + bef,
// scatter-added into agg[dst] with native fp32 global atomics. Wef/bef in
// LDS. m1/agg (25.6 MB each) stay resident in the 192 MB L2.
// ---------------------------------------------------------------------------
__global__ __launch_bounds__(256) void edge_message_scatter(
    const float* __restrict__ m1, const float* __restrict__ edge_attr,
    const int* __restrict__ src, const int* __restrict__ dst,
    const float* __restrict__ Wef, const float* __restrict__ bef,
    float* __restrict__ agg)
{
    __shared__ float sW[EDIM * HDIM];
    __shared__ float sB[HDIM];
    const int tid = threadIdx.x;
    for (int i = tid; i < EDIM * HDIM; i += 256) sW[i] = Wef[i];
    if (tid < HDIM) sB[tid] = bef[tid];
    __syncthreads();

    const int lane = tid & 31;
    const int wv   = tid >> 5;
    const int e    = blockIdx.x * 8 + wv;
    if (e >= N_EDGES) return;

    const int s = src[e];
    const int d = dst[e];
    const float4 ea = *(const float4*)(edge_attr + (size_t)e * EDIM);
    const int c0 = lane * 4;
    const float4 mv = *(const float4*)(m1 + (size_t)s * HDIM + c0);
    const float vals[4] = {mv.x, mv.y, mv.z, mv.w};
    float* aggp = agg + (size_t)d * HDIM + c0;

    #pragma unroll
    for (int j = 0; j < 4; ++j) {
        const int c = c0 + j;
        const float v = vals[j]
            + ea.x * sW[0 * HDIM + c] + ea.y * sW[1 * HDIM + c]
            + ea.z * sW[2 * HDIM + c] + ea.w * sW[3 * HDIM + c]
            + sB[c];
        atomic_add_f32_dev(aggp + j, v);
    }
}

// ---------------------------------------------------------------------------
// LayerNorm + ReLU residual: one wave32 per node, __shfl_xor reductions.
// ---------------------------------------------------------------------------
__global__ __launch_bounds__(256) void layernorm_residual(
    const float* __restrict__ agg, const float* __restrict__ gamma,
    const float* __restrict__ beta, float* __restrict__ h)
{
    const int tid  = threadIdx.x;
    const int lane = tid & 31;
    const int wv   = tid >> 5;
    const int node = blockIdx.x * 8 + wv;
    if (node >= N_NODES) return;

    const int c0 = lane * 4;
    const float4 a = *(const float4*)(agg + (size_t)node * HDIM + c0);
    float s  = a.x + a.y + a.z + a.w;
    float sq = a.x * a.x + a.y * a.y + a.z * a.z + a.w * a.w;
    #pragma unroll
    for (int off = 16; off > 0; off >>= 1) {
        s  += __shfl_xor(s, off, 32);
        sq += __shfl_xor(sq, off, 32);
    }
    const float mean = s * (1.0f / HDIM);
    const float var  = sq * (1.0f / HDIM) - mean * mean;
    const float inv  = rsqrtf(var + LN_EPS);

    const float av[4] = {a.x, a.y, a.z, a.w};
    float4 hv = *(float4*)(h + (size_t)node * HDIM + c0);
    float out[4] = {hv.x, hv.y, hv.z, hv.w};
    #pragma unroll
    for (int j = 0; j < 4; ++j) {
        const int c = c0 + j;
        const float xn = (av[j] - mean) * inv * gamma[c] + beta[c];
        out[j] += fmaxf(xn, 0.0f);
    }
    *(float4*)(h + (size_t)node * HDIM + c0) =
        make_float4(out[0], out[1], out[2], out[3]);
}

// ---------------------------------------------------------------------------
extern "C" void kernel_launch(void* const* d_in, const int* in_sizes, int n_in,
                              void* d_out, int out_size, void* d_ws, size_t ws_size,
                              hipStream_t stream)
{
    const float* x          = (const float*)d_in[0];
    const int*   edge_index = (const int*)d_in[1];
    const float* edge_attr  = (const float*)d_in[2];
    const float* Wp         = (const float*)d_in[3];
    const float* bp         = (const float*)d_in[4];
    const float* Wn         = (const float*)d_in[5];
    const float* bn         = (const float*)d_in[6];
    const float* We         = (const float*)d_in[7];
    const float* be         = (const float*)d_in[8];
    const float* Wm         = (const float*)d_in[9];
    const float* bm         = (const float*)d_in[10];
    const float* gamma      = (const float*)d_in[11];
    const float* beta       = (const float*)d_in[12];
    float* h = (float*)d_out;

    // Workspace layout (floats): m1 | agg | WfusedT | bfused | Wefused |
    // befused | WpT   (~51.4 MB total)
    float* ws       = (float*)d_ws;
    float* m1       = ws;
    float* agg      = m1 + (size_t)N_NODES * HDIM;
    float* WfusedT  = agg + (size_t)N_NODES * HDIM;
    float* bfused   = WfusedT + (size_t)NLAYERS * HDIM * HDIM;
    float* Wefused  = bfused + (size_t)NLAYERS * HDIM;
    float* befused  = Wefused + (size_t)NLAYERS * EDIM * HDIM;
    float* WpT      = befused + (size_t)NLAYERS * HDIM;

    const int* src = edge_index;            // edge_index[0]
    const int* dst = edge_index + N_EDGES;  // edge_index[1]

    // 1) fuse / transpose weights (tiny)
    {
        int t = NLAYERS * HDIM * HDIM;
        fuse_node_weights<<<(t + 255) / 256, 256, 0, stream>>>(Wn, Wm, WfusedT);
    }
    {
        int t = NDIM * HDIM;
        transpose_wp<<<(t + 255) / 256, 256, 0, stream>>>(Wp, WpT);
    }
    {
        int t = NLAYERS * EDIM * HDIM;
        fuse_edge_weights<<<(t + 255) / 256, 256, 0, stream>>>(We, Wm, Wefused);
    }
    {
        int t = NLAYERS * HDIM;
        fuse_biases<<<(t + 255) / 256, 256, 0, stream>>>(bn, be, bm, Wm, bfused, befused);
    }

    // 2) input projection: h = x @ Wp + bp   (K = 32, transposed weights)
    gemm_wmma_f32<<<N_NODES / 16, 256, 0, stream>>>(x, WpT, bp, h, NDIM);

    // 3) layers
    const int n4 = (N_NODES * HDIM) / 4;
    for (int l = 0; l < NLAYERS; ++l) {
        gemm_wmma_f32<<<N_NODES / 16, 256, 0, stream>>>(
            h, WfusedT + (size_t)l * HDIM * HDIM, bfused + l * HDIM, m1, HDIM);

        zero_f32v4<<<(n4 + 255) / 256, 256, 0, stream>>>((float4*)agg, n4);

        edge_message_scatter<<<(N_EDGES + 7) / 8, 256, 0, stream>>>(
            m1, edge_attr, src, dst,
            Wefused + (size_t)l * EDIM * HDIM, befused + l * HDIM, agg);

        layernorm_residual<<<(N_NODES + 7) / 8, 256, 0, stream>>>(
            agg, gamma + l * HDIM, beta + l * HDIM, h);
    }
}